// FFTConvReservoir_81595788689495
// MI455X (gfx1250) — compile-verified
//
#include <hip/hip_runtime.h>
#include <math.h>

// FFT-based long convolution + skip add, fused single kernel for MI455X.
//   y = ifft( fft(u,N) * fft(K,N) )[:L] + D*u,  N = 16384, L = 8192
// Design: one workgroup per (b,h) row; whole transform lives in LDS
// (2 x 128KB complex Stockham ping-pong + 32KB u stash = 288KB of the WGP's
// 320KB). Real-packing (z = u + iK) gives both spectra from ONE forward FFT.
// Radix-16 stages are 16x16 complex matmuls by the DFT-16 matrix, executed
// with chained V_WMMA_F32_16X16X4_F32 (fp32 WMMA = correct precision path).

typedef __attribute__((ext_vector_type(2))) float v2f;
typedef __attribute__((ext_vector_type(8))) float v8f;

#define NFFT 16384
#define LSEQ 8192
#define TPB  256
#define PI2  6.28318530717958647692f

// D = A(16x16) * B(16x16) + acc, via 4 chained 16x16x4 f32 WMMAs.
// A[j], B[j] (j=0..7) hold the lane's elements in the ISA A/B layout:
//   slice kk = j>>1, vgpr = j&1, K = kk*4 + (j&1) + (lane>=16 ? 2 : 0).
static __device__ __forceinline__ v8f wmma16(const float* A, const float* B, v8f acc) {
#pragma unroll
  for (int kk = 0; kk < 4; ++kk) {
    v2f a; a.x = A[2 * kk]; a.y = A[2 * kk + 1];
    v2f b; b.x = B[2 * kk]; b.y = B[2 * kk + 1];
    acc = __builtin_amdgcn_wmma_f32_16x16x4_f32(false, a, false, b, (short)0, acc,
                                                false, false);
  }
  return acc;
}

// One Stockham radix-16 stage (out-of-place, src -> dst).
//   reads  x[q + s*(p + k*(n/16))], k = 0..15
//   writes y[q + s*(16p + u)] * W_n^(tsgn * p*u),  u = 0..15
// 64 butter 16-groups per batch-of-16 -> 8 batches per wave (8 waves).
// Cm = cos(2pi mk/16); S1/S2 = +/-sin picked per direction:
//   forward:  Yr = C*Br + S*Bi   ; Yi = C*Bi + (-S)*Br
//   inverse:  Yr = C*Br + (-S)*Bi; Yi = C*Bi +   S *Br
static __device__ __forceinline__ void r16_stage(
    const float* __restrict__ sr, const float* __restrict__ si,
    float* __restrict__ dr_, float* __restrict__ di_,
    int n, int ls, float tsgn,
    const float* Cm, const float* S1, const float* S2, int lane, int wav) {
  const int s   = 1 << ls;
  const int nr  = n >> 4;
  const int hi2 = (lane >> 4) << 1;   // K half-wave offset {0,2}
  const int col = lane & 15;          // B/D column (N index)
  const int mb  = (lane >> 4) << 3;   // D row offset {0,8}
#pragma unroll 1
  for (int it = 0; it < 8; ++it) {
    const int t = (((wav << 3) + it) << 4) + col;   // butterfly id for this column
    const int p = t >> ls;
    const int q = t & (s - 1);
    float Br[8], Bi[8];
#pragma unroll
    for (int j = 0; j < 8; ++j) {
      const int k   = ((j >> 1) << 2) + (j & 1) + hi2;
      const int idx = q + ((p + k * nr) << ls);
      Br[j] = sr[idx];
      Bi[j] = si[idx];
    }
    v8f yr = {0.f, 0.f, 0.f, 0.f, 0.f, 0.f, 0.f, 0.f};
    v8f yi = {0.f, 0.f, 0.f, 0.f, 0.f, 0.f, 0.f, 0.f};
    yr = wmma16(Cm, Br, yr);
    yr = wmma16(S1, Bi, yr);
    yi = wmma16(Cm, Bi, yi);
    yi = wmma16(S2, Br, yi);
    const float angb = tsgn * (PI2 / (float)n) * (float)p;
#pragma unroll
    for (int rv = 0; rv < 8; ++rv) {
      const int   uu  = rv + mb;
      const float ang = angb * (float)uu;
      const float wr = __cosf(ang), wi = __sinf(ang);
      const float a = yr[rv], b = yi[rv];
      const int odx = q + (((p << 4) + uu) << ls);
      dr_[odx] = a * wr - b * wi;
      di_[odx] = a * wi + b * wr;
    }
  }
}

// Final radix-4 stage: n=4, s=4096, p=0 -> unit twiddles, in-place.
// sg = +1 forward, -1 inverse.
static __device__ __forceinline__ void r4_stage(float* br, float* bi, float sg, int tid) {
#pragma unroll 1
  for (int q = tid; q < 4096; q += TPB) {
    const float a0r = br[q],          a0i = bi[q];
    const float a1r = br[q + 4096],   a1i = bi[q + 4096];
    const float a2r = br[q + 8192],   a2i = bi[q + 8192];
    const float a3r = br[q + 12288],  a3i = bi[q + 12288];
    const float s02r = a0r + a2r, s02i = a0i + a2i;
    const float d02r = a0r - a2r, d02i = a0i - a2i;
    const float s13r = a1r + a3r, s13i = a1i + a3i;
    const float d13r = a1r - a3r, d13i = a1i - a3i;
    br[q]         = s02r + s13r;      bi[q]         = s02i + s13i;
    br[q + 8192]  = s02r - s13r;      bi[q + 8192]  = s02i - s13i;
    br[q + 4096]  = d02r + sg * d13i; bi[q + 4096]  = d02i - sg * d13r;
    br[q + 12288] = d02r - sg * d13i; bi[q + 12288] = d02i + sg * d13r;
  }
}

__global__ void __launch_bounds__(TPB)
fftconv_reservoir(const float* __restrict__ u, const float* __restrict__ Kk,
                  const float* __restrict__ Dd, float* __restrict__ out, int H) {
  extern __shared__ float smem[];
  float* Xr = smem;              // 16384
  float* Xi = Xr + NFFT;         // 16384
  float* Yr = Xi + NFFT;         // 16384
  float* Yi = Yr + NFFT;         // 16384
  float* Us = Yi + NFFT;         // 8192 (u stash for skip add)

  const int row  = blockIdx.x;         // b*H + h
  const int h    = row % H;
  const int tid  = threadIdx.x;
  const int lane = tid & 31;
  const int wav  = tid >> 5;

  const float4* u4 = (const float4*)(u  + (size_t)row * LSEQ);
  const float4* k4 = (const float4*)(Kk + (size_t)h   * LSEQ);

  // ---- load: z = u + i*K, zero-pad to NFFT, stash u ----
  for (int i = tid; i < LSEQ / 4; i += TPB) {
    const float4 a = u4[i];
    const float4 b = k4[i];
    const int j = i << 2;
    Xr[j] = a.x; Xr[j + 1] = a.y; Xr[j + 2] = a.z; Xr[j + 3] = a.w;
    Xi[j] = b.x; Xi[j + 1] = b.y; Xi[j + 2] = b.z; Xi[j + 3] = b.w;
    Us[j] = a.x; Us[j + 1] = a.y; Us[j + 2] = a.z; Us[j + 3] = a.w;
  }
  for (int j = LSEQ + tid; j < NFFT; j += TPB) { Xr[j] = 0.f; Xi[j] = 0.f; }

  // ---- per-lane DFT-16 matrices in WMMA A layout ----
  float Cm[8], Sm[8], NSm[8];
  {
    const int hi2 = (lane >> 4) << 1;
    const int m   = lane & 15;
#pragma unroll
    for (int j = 0; j < 8; ++j) {
      const int k = ((j >> 1) << 2) + (j & 1) + hi2;
      const float th = (PI2 / 16.f) * (float)(m * k);
      Cm[j] = __cosf(th);
      Sm[j] = __sinf(th);
      NSm[j] = -Sm[j];
    }
  }
  __syncthreads();

  // ---- forward FFT: X -> Y -> X -> Y, radix-4 in place on Y ----
  r16_stage(Xr, Xi, Yr, Yi, 16384, 0, -1.f, Cm, Sm, NSm, lane, wav); __syncthreads();
  r16_stage(Yr, Yi, Xr, Xi,  1024, 4, -1.f, Cm, Sm, NSm, lane, wav); __syncthreads();
  r16_stage(Xr, Xi, Yr, Yi,    64, 8, -1.f, Cm, Sm, NSm, lane, wav); __syncthreads();
  r4_stage(Yr, Yi, 1.f, tid);                                        __syncthreads();

  // ---- unpack real-packed spectra and multiply: P = U_s * K_s (into X) ----
  // Z = fft(u + iK); U(f) = (Z(f)+conj(Z(-f)))/2 ; K(f) = (Z(f)-conj(Z(-f)))/(2i)
  for (int f = tid; f <= NFFT / 2; f += TPB) {
    const int g = (NFFT - f) & (NFFT - 1);
    const float a = Yr[f], b = Yi[f], c = Yr[g], d = Yi[g];
    const float ur = 0.5f * (a + c), ui = 0.5f * (b - d);
    const float kr = 0.5f * (b + d), ki = -0.5f * (a - c);
    const float pr = ur * kr - ui * ki;
    const float pi = ur * ki + ui * kr;
    Xr[f] = pr; Xi[f] = pi;
    Xr[g] = pr; Xi[g] = -pi;   // conjugate symmetry of real*real spectrum
  }
  __syncthreads();

  // ---- inverse FFT: X -> Y -> X -> Y, radix-4 in place on Y ----
  r16_stage(Xr, Xi, Yr, Yi, 16384, 0, 1.f, Cm, NSm, Sm, lane, wav); __syncthreads();
  r16_stage(Yr, Yi, Xr, Xi,  1024, 4, 1.f, Cm, NSm, Sm, lane, wav); __syncthreads();
  r16_stage(Xr, Xi, Yr, Yi,    64, 8, 1.f, Cm, NSm, Sm, lane, wav); __syncthreads();
  r4_stage(Yr, Yi, -1.f, tid);                                      __syncthreads();

  // ---- scale 1/N, skip add D*u, store complex64 (interleaved float2) ----
  const float inv = 1.0f / (float)NFFT;
  const float Dv  = Dd[h];
  float2* o2 = (float2*)out + (size_t)row * LSEQ;
  for (int l = tid; l < LSEQ; l += TPB) {
    float2 v;
    v.x = Yr[l] * inv + Dv * Us[l];
    v.y = Yi[l] * inv;
    o2[l] = v;
  }
}

extern "C" void kernel_launch(void* const* d_in, const int* in_sizes, int n_in,
                              void* d_out, int out_size, void* d_ws, size_t ws_size,
                              hipStream_t stream) {
  (void)n_in; (void)d_ws; (void)ws_size; (void)out_size;
  const float* u  = (const float*)d_in[0];  // (B,H,L) f32
  const float* K  = (const float*)d_in[1];  // (H,L)   f32
  const float* D  = (const float*)d_in[2];  // (H,)    f32
  float* out = (float*)d_out;               // (B,H,L) complex64, interleaved

  const int H    = in_sizes[2];             // 256
  const int rows = in_sizes[0] / LSEQ;      // B*H = 2048

  const size_t shmem = (size_t)(4 * NFFT + LSEQ) * sizeof(float); // 288 KB
  fftconv_reservoir<<<rows, TPB, shmem, stream>>>(u, K, D, out, H);
}